// RBFNN_43319040148016
// MI455X (gfx1250) — compile-verified
//
#include <hip/hip_runtime.h>
#include <cmath>

typedef __attribute__((ext_vector_type(2))) float v2f;
typedef __attribute__((ext_vector_type(4))) float v4f;
typedef __attribute__((ext_vector_type(8))) float v8f;

#define Bn   16384
#define Fn   512
#define Cn   1024
#define Kn   100
#define KT   7         // 7 tiles of 16 -> 112 >= 100
#define KP   112
#define MBLK 64        // rows per block
#define NWAVE 4        // wave32 per block
#define CBLK 32        // centers per outer iteration (2 N-tiles per wave)
#define CSP  516       // centers LDS pitch; 516 mod 64 = 4 -> conflict free, 16B rows
#define WSP  36        // W LDS pitch; 36k distinct mod 64 for k<16; 144B rows (16B aligned)
#define PHP  36        // phi LDS pitch (row = 32 phis + pad), even -> 8B aligned

// dynamic LDS partition (floats)
#define OFF_CS   0
#define OFF_WS   (OFF_CS + CBLK * CSP)          // 16512
#define OFF_PH   (OFF_WS + KP * WSP)            // 20544
#define OFF_XSQ  (OFF_PH + NWAVE * 16 * PHP)    // 22848
#define OFF_CSQ  (OFF_XSQ + MBLK)               // 22912
#define SMEM_FLOATS (OFF_CSQ + CBLK)            // 22944
#define SMEM_BYTES  (SMEM_FLOATS * 4)           // 91776 B (< 320KB/WGP, ~3 blocks resident)

__launch_bounds__(128, 1)
__global__ void rbf_fused_kernel(const float* __restrict__ x,
                                 const float* __restrict__ centers,
                                 const float* __restrict__ gammap,
                                 const float* __restrict__ W,
                                 const float* __restrict__ bias,
                                 float* __restrict__ out) {
    extern __shared__ __attribute__((aligned(16))) float smem[];
    float* cs   = smem + OFF_CS;
    float* ws   = smem + OFF_WS;
    float* phis = smem + OFF_PH;
    float* xsq  = smem + OFF_XSQ;
    float* csq  = smem + OFF_CSQ;

    const int tid  = threadIdx.x;
    const int wave = tid >> 5;
    const int lane = tid & 31;
    const int lh   = lane >> 4;   // lane-half
    const int l16  = lane & 15;
    const int blockRow = blockIdx.x * MBLK;
    const int m0 = blockRow + wave * 16;
    const float g = gammap[0];

    // ---- per-row ||x||^2 (once per block) ----
    if (tid < MBLK) {
        const float* xr = x + (size_t)(blockRow + tid) * Fn;
        float s = 0.f;
        for (int f = 0; f < Fn; f += 4) {
            v4f v = *(const v4f*)(xr + f);
            s += v[0]*v[0] + v[1]*v[1] + v[2]*v[2] + v[3]*v[3];
        }
        xsq[tid] = s;
    }

    v8f acc[KT];
    #pragma unroll
    for (int t = 0; t < KT; ++t)
        #pragma unroll
        for (int i = 0; i < 8; ++i) acc[t][i] = 0.f;

    float* ph = &phis[wave * 16 * PHP];
    const float* xa  = x + (size_t)(m0 + l16) * Fn + 2 * lh;  // A-frag base (GEMM-1)
    const float* cb0 = &cs[(l16     ) * CSP + 2 * lh];        // B-frag, N-tile 0
    const float* cb1 = &cs[(l16 + 16) * CSP + 2 * lh];        // B-frag, N-tile 1

    for (int c0 = 0; c0 < Cn; c0 += CBLK) {
        __syncthreads();   // protect previous iteration's cs/ws/csq readers

        if (tid < CBLK) csq[tid] = 0.f;

        // ---- stage 32 centers rows into LDS + prefetch next chunk ----
        for (int idx = tid; idx < CBLK * (Fn / 4); idx += 128) {
            int c  = idx >> 7;       // Fn/4 == 128 float4 per row
            int fq = idx & 127;
            const float* src = centers + (size_t)(c0 + c) * Fn + 4 * fq;
            v4f v = *(const v4f*)src;
            *(v4f*)(&cs[c * CSP + 4 * fq]) = v;
            if (c0 + CBLK < Cn)
                __builtin_prefetch(src + (size_t)CBLK * Fn, 0, 1);
        }
        // ---- stage W chunk [112 x 32] into LDS (k >= 100 zero-filled) ----
        for (int idx = tid; idx < KP * (CBLK / 4); idx += 128) {
            int k  = idx >> 3;       // CBLK/4 == 8 float4 per row
            int cq = idx & 7;
            v4f v; v[0] = 0.f; v[1] = 0.f; v[2] = 0.f; v[3] = 0.f;
            if (k < Kn) v = *(const v4f*)(W + (size_t)k * Cn + c0 + 4 * cq);
            *(v4f*)(&ws[k * WSP + 4 * cq]) = v;
        }
        __syncthreads();

        // ---- ||c||^2: 4 threads per center, combine via LDS float atomics ----
        {
            int c = tid >> 2, q = tid & 3;
            const float* cr = &cs[c * CSP + q * 128];
            float s = 0.f;
            for (int f = 0; f < 128; f += 4) {
                v4f v = *(const v4f*)(cr + f);
                s += v[0]*v[0] + v[1]*v[1] + v[2]*v[2] + v[3]*v[3];
            }
            atomicAdd(&csq[c], s);   // ds_add_f32
        }
        __syncthreads();

        // ---- GEMM-1: two independent 16x16 cross tiles, K-loop over F ----
        v8f cross0, cross1;
        #pragma unroll
        for (int i = 0; i < 8; ++i) { cross0[i] = 0.f; cross1[i] = 0.f; }

        #pragma unroll 4
        for (int f0 = 0; f0 < Fn; f0 += 4) {
            v2f a  = *(const v2f*)(xa  + f0);
            v2f b0 = *(const v2f*)(cb0 + f0);
            v2f b1 = *(const v2f*)(cb1 + f0);
            cross0 = __builtin_amdgcn_wmma_f32_16x16x4_f32(
                false, a, false, b0, (short)0, cross0, false, false);
            cross1 = __builtin_amdgcn_wmma_f32_16x16x4_f32(
                false, a, false, b1, (short)0, cross1, false, false);
        }

        // ---- phi = exp(-g*sqrt(max(xsq+csq-2*cross,0))) -> per-wave LDS tile ----
        // raw v_sqrt_f32 (1-ulp) instead of the IEEE-refined sequence: plenty for exp()
        {
            const float cq0 = csq[l16];
            const float cq1 = csq[l16 + 16];
            #pragma unroll
            for (int i = 0; i < 8; ++i) {
                int mrow = i + 8 * lh;                 // C/D layout row
                float xq = xsq[wave * 16 + mrow];
                float d2a = fmaxf(xq + cq0 - 2.0f * cross0[i], 0.f);
                float d2b = fmaxf(xq + cq1 - 2.0f * cross1[i], 0.f);
                float da = __builtin_amdgcn_sqrtf(d2a);   // v_sqrt_f32
                float db = __builtin_amdgcn_sqrtf(d2b);   // v_sqrt_f32
                ph[mrow * PHP + l16     ] = __expf(-g * da);
                ph[mrow * PHP + l16 + 16] = __expf(-g * db);
            }
        }

        // ---- GEMM-2: acc[m,k] += phi[m,c] * W[k,c] over 32-c chunk ----
        #pragma unroll
        for (int j = 0; j < 8; ++j) {
            v2f a2 = *(const v2f*)(&ph[l16 * PHP + 4 * j + 2 * lh]);
            #pragma unroll
            for (int t = 0; t < KT; ++t) {
                v2f b2 = *(const v2f*)(&ws[(t * 16 + l16) * WSP + 4 * j + 2 * lh]);
                acc[t] = __builtin_amdgcn_wmma_f32_16x16x4_f32(
                    false, a2, false, b2, (short)0, acc[t], false, false);
            }
        }
    }

    // ---- epilogue: add bias, store [B, 100] ----
    #pragma unroll
    for (int t = 0; t < KT; ++t) {
        int kk = t * 16 + l16;
        if (kk < Kn) {
            float bb = bias[kk];
            #pragma unroll
            for (int i = 0; i < 8; ++i) {
                int r = m0 + i + 8 * lh;
                out[(size_t)r * Kn + kk] = acc[t][i] + bb;
            }
        }
    }
}

extern "C" void kernel_launch(void* const* d_in, const int* in_sizes, int n_in,
                              void* d_out, int out_size, void* d_ws, size_t ws_size,
                              hipStream_t stream) {
    (void)in_sizes; (void)n_in; (void)d_ws; (void)ws_size; (void)out_size;
    const float* x       = (const float*)d_in[0];
    const float* centers = (const float*)d_in[1];
    const float* gammap  = (const float*)d_in[2];
    const float* W       = (const float*)d_in[3];
    const float* b       = (const float*)d_in[4];
    float* out = (float*)d_out;

    dim3 grid(Bn / MBLK);   // 256 workgroups
    dim3 block(128);        // 4 wave32
    rbf_fused_kernel<<<grid, block, SMEM_BYTES, stream>>>(x, centers, gammap, W, b, out);
}